// EpisodicCountModule_37082747634611
// MI455X (gfx1250) — compile-verified
//
#include <hip/hip_runtime.h>
#include <hip/hip_bf16.h>
#include <math.h>

typedef __attribute__((ext_vector_type(2))) float v2f;
typedef __attribute__((ext_vector_type(8))) float v8f;

#define BATCH 64
#define SEQ   1024
#define FEAT  512
#define NBINS 32
#define NROWS (BATCH * SEQ)          // 65536
#define CHUNKS 128
#define ROWS_PER_CHUNK (NROWS / CHUNKS)  // 512
#define SP_STRIDE 516                // FEAT + 4: stride%64==4 -> conflict-free B reads

// ---------------------------------------------------------------------------
// Kernel 1a: per-chunk partial sum / sum-of-squares per feature (deterministic)
// grid = CHUNKS blocks of FEAT threads; thread f reads column f of its chunk.
// Consecutive threads -> consecutive addresses within a row (coalesced 2KB).
// ---------------------------------------------------------------------------
__global__ void __launch_bounds__(FEAT)
k_partials(const float* __restrict__ x, float* __restrict__ ps, float* __restrict__ pq) {
    const int f = threadIdx.x;
    const int b = blockIdx.x;
    const float* p = x + (size_t)b * ROWS_PER_CHUNK * FEAT + f;
    float s = 0.f, q = 0.f;
#pragma unroll 4
    for (int r = 0; r < ROWS_PER_CHUNK; ++r) {
        float v = p[(size_t)r * FEAT];
        s += v;
        q += v * v;
    }
    ps[b * FEAT + f] = s;
    pq[b * FEAT + f] = q;
}

// ---------------------------------------------------------------------------
// Kernel 1b: finalize RunningMeanStd update (fresh state, count=1e-4)
// mean[f], invsigma[f] per feature. 1 block x 512 threads, fixed order sums.
// ---------------------------------------------------------------------------
__global__ void __launch_bounds__(FEAT)
k_stats(const float* __restrict__ ps, const float* __restrict__ pq,
        float* __restrict__ meanO, float* __restrict__ invsO) {
    const int f = threadIdx.x;
    float s = 0.f, q = 0.f;
    for (int b = 0; b < CHUNKS; ++b) {
        s += ps[b * FEAT + f];
        q += pq[b * FEAT + f];
    }
    const float N = (float)NROWS;
    const float bm   = s / N;
    const float bvar = (q - N * bm * bm) / (N - 1.0f);   // unbiased
    const float tot  = N + 1e-4f;
    const float mean = bm * N / tot;
    const float M2   = 1e-4f + bvar * N + bm * bm * 1e-4f * N / tot;
    const float var  = M2 / tot;
    meanO[f] = mean;
    invsO[f] = 1.0f / sqrtf(var + 1e-8f);
}

// ---------------------------------------------------------------------------
// Kernel 2: fold normalization into projection. Pp is written TRANSPOSED
// ([NBINS][FEAT], row-major) so the GEMM kernel's per-lane B fragment
// {Pp[kk][n], Pp[kk+1][n]} is an 8-byte contiguous run in LDS.
//   PpT[n*FEAT + f] = P[f,n] * invsigma[f]
//   bias[n]         = sum_f mean[f]*invsigma[f]*P[f,n]
// ---------------------------------------------------------------------------
__global__ void __launch_bounds__(256)
k_proj(const float* __restrict__ P, const float* __restrict__ mean,
       const float* __restrict__ invs, float* __restrict__ PpT, float* __restrict__ bias) {
    const int tid = threadIdx.x;
    for (int e = tid; e < FEAT * NBINS; e += 256) {
        const int f = e >> 5;
        const int n = e & 31;
        PpT[n * FEAT + f] = P[e] * invs[f];
    }
    if (tid < NBINS) {
        float acc = 0.f;
        for (int f = 0; f < FEAT; ++f)
            acc += mean[f] * invs[f] * P[f * NBINS + tid];
        bias[tid] = acc;
    }
}

// ---------------------------------------------------------------------------
// Kernel 3: WMMA GEMM  x[65536x512] @ Pp[512x32], sign -> 32-bit key per row.
// One wave owns a 16-row x 32-bin tile: two v8f accumulators, K stepped by 4
// with V_WMMA_F32_16X16X4_F32.  PpT staged in LDS (transposed, stride 516:
// 516 % 64 == 4 so lanes 0..15 read banks {4l+kk, 4l+kk+1} -> conflict-free,
// and each B fragment is one ds_load_b64, no repack movs).
//
// f32 A 16x4 layout: lanes 0-15 = rows M, VGPR0/1 = K{0,1} (lanes<16) or
// K{2,3} (lanes>=16).  B 4x16 mirrored.  C/D: VGPR j = row j (lanes<16) /
// row j+8 (lanes>=16), N = lane&15 -> ballot assembles key bits directly.
// ---------------------------------------------------------------------------
__global__ void __launch_bounds__(256)
k_gemm_keys(const float* __restrict__ x, const float* __restrict__ PpT,
            const float* __restrict__ bias, unsigned int* __restrict__ keys) {
    __shared__ float sP[NBINS * SP_STRIDE];     // ~64.5 KB, padded stride
    const int tid = threadIdx.x;
    for (int e = tid; e < NBINS * FEAT; e += 256) {
        const int n = e >> 9;        // e / FEAT
        const int f = e & 511;       // e % FEAT
        sP[n * SP_STRIDE + f] = PpT[e];
    }
    __syncthreads();

    const int wave  = tid >> 5;
    const int lane  = tid & 31;
    const int lrow  = lane & 15;
    const int khalf = lane >> 4;                // 0: K0..1, 1: K2..3
    const int tile  = blockIdx.x * 8 + wave;    // 0..4095
    const int m0    = tile * 16;

    const float* ap  = x + (size_t)(m0 + lrow) * FEAT + khalf * 2;
    const float* bp0 = sP + lrow * SP_STRIDE + khalf * 2;          // bins 0..15
    const float* bp1 = sP + (lrow + 16) * SP_STRIDE + khalf * 2;   // bins 16..31

    v8f acc0 = {};
    v8f acc1 = {};
    for (int k = 0; k < FEAT; k += 4) {
        v2f a, b0, b1;
        a.x  = ap[k];      a.y  = ap[k + 1];
        b0.x = bp0[k];     b0.y = bp0[k + 1];
        b1.x = bp1[k];     b1.y = bp1[k + 1];
        acc0 = __builtin_amdgcn_wmma_f32_16x16x4_f32(false, a, false, b0,
                                                     (short)0, acc0, false, false);
        acc1 = __builtin_amdgcn_wmma_f32_16x16x4_f32(false, a, false, b1,
                                                     (short)0, acc1, false, false);
    }

    const float blo = bias[lrow];        // bins 0..15  (N = lane&15)
    const float bhi = bias[lrow + 16];   // bins 16..31
    unsigned int mykey = 0;
#pragma unroll
    for (int j = 0; j < 8; ++j) {
        unsigned int p0 = __builtin_amdgcn_ballot_w32(acc0[j] > blo);
        unsigned int p1 = __builtin_amdgcn_ballot_w32(acc1[j] > bhi);
        unsigned int keyLo = (p0 & 0xFFFFu) | ((p1 & 0xFFFFu) << 16);  // row j
        unsigned int keyHi = (p0 >> 16)     | (p1 & 0xFFFF0000u);      // row j+8
        if ((lane & 7) == j)
            mykey = (lane & 8) ? keyHi : keyLo;
    }
    if (lane < 16)
        keys[m0 + lane] = mykey;
}

// ---------------------------------------------------------------------------
// Kernel 4: ordered occurrence counts. env = i % 64, so class e holds rows
// i = e + 64*t, t = 0..1023 in sample order. One block per class; keys in LDS;
// thread t counts matches among t' <= t; reward = 1/sqrt(count).
// ---------------------------------------------------------------------------
__global__ void __launch_bounds__(SEQ)
k_count(const unsigned int* __restrict__ keys, float* __restrict__ out) {
    __shared__ unsigned int sk[SEQ];
    const int e = blockIdx.x;        // env class
    const int t = threadIdx.x;       // order within class
    sk[t] = keys[e + BATCH * t];
    __syncthreads();
    const unsigned int myk = sk[t];
    int c = 0;
    for (int tp = 0; tp <= t; ++tp)
        c += (sk[tp] == myk) ? 1 : 0;
    out[e + BATCH * t] = rsqrtf((float)c);
}

// ---------------------------------------------------------------------------
extern "C" void kernel_launch(void* const* d_in, const int* in_sizes, int n_in,
                              void* d_out, int out_size, void* d_ws, size_t ws_size,
                              hipStream_t stream) {
    const float* x = (const float*)d_in[0];   // features [64,1024,512]
    const float* P = (const float*)d_in[1];   // random_projection [512,32]
    float* out = (float*)d_out;               // rewards [64,1024,1] f32

    float* ws   = (float*)d_ws;
    float* ps   = ws;                               // CHUNKS*FEAT
    float* pq   = ps + CHUNKS * FEAT;               // CHUNKS*FEAT
    float* mean = pq + CHUNKS * FEAT;               // FEAT
    float* invs = mean + FEAT;                      // FEAT
    float* PpT  = invs + FEAT;                      // NBINS*FEAT (transposed)
    float* bias = PpT + NBINS * FEAT;               // NBINS
    unsigned int* keys = (unsigned int*)(bias + NBINS);  // NROWS u32

    k_partials<<<CHUNKS, FEAT, 0, stream>>>(x, ps, pq);
    k_stats<<<1, FEAT, 0, stream>>>(ps, pq, mean, invs);
    k_proj<<<1, 256, 0, stream>>>(P, mean, invs, PpT, bias);
    k_gemm_keys<<<NROWS / 16 / 8, 256, 0, stream>>>(x, PpT, bias, keys);
    k_count<<<BATCH, SEQ, 0, stream>>>(keys, out);
}